// MoEGate_1108101562792
// MI455X (gfx1250) — compile-verified
//
#include <hip/hip_runtime.h>
#include <math.h>

typedef float v2f __attribute__((ext_vector_type(2)));
typedef float v8f __attribute__((ext_vector_type(8)));

#define T_TOKENS   16384
#define BATCH      4
#define SEQ        4096
#define DIM        2048
#define NEXP       64
#define TOPK       8
#define KC         128           // K-chunk staged in LDS
#define KSTRIDE    (KC + 4)      // 528B row stride: 16B-aligned for async B128,
                                 // and (4m+c)%64 banks -> conflict-free b64 frags
#define TOK_PER_WG 64
#define LSTRIDE    (NEXP + 1)    // padded logits row in LDS
#define ALPHA_F    1e-3f

// ---------------------------------------------------------------------------
// Fused MoE gate: fp32 WMMA GEMM + softmax + top-8 + aux-loss partials.
// Staging uses GLOBAL_LOAD_ASYNC_TO_LDS_B128 (ASYNCcnt) with double-buffered
// LDS tiles so HBM latency overlaps the WMMA stream.
// grid = 256 WGs, block = 128 threads (4 wave32)
// ---------------------------------------------------------------------------
__global__ __launch_bounds__(128) void moe_gate_kernel(
    const float* __restrict__ x,        // [T, DIM]
    const float* __restrict__ w,        // [NEXP, DIM]
    float* __restrict__ out_idx,        // [T, TOPK] (indices stored as float)
    float* __restrict__ out_w,          // [T, TOPK]
    float* __restrict__ g_score_sum,    // [BATCH, NEXP] (atomic accum)
    float* __restrict__ g_count)        // [BATCH, NEXP] (atomic accum)
{
    __shared__ float xs[2][TOK_PER_WG * KSTRIDE];   // 2 x 33,792 B
    __shared__ float ws2[2][NEXP * KSTRIDE];        // 2 x 33,792 B
    __shared__ float lprob[TOK_PER_WG * LSTRIDE];   // 16,640 B
    __shared__ unsigned int lcount[NEXP];

    const int tid   = threadIdx.x;        // 0..127
    const int wave  = tid >> 5;           // 0..3
    const int lane  = tid & 31;
    const int half  = lane >> 4;          // 0/1 (K-half of the fragment)
    const int row16 = lane & 15;          // M (A) / N (B,C) index within tile
    const int tok0  = blockIdx.x * TOK_PER_WG;

    if (tid < NEXP) lcount[tid] = 0u;

    // ---- async stage of one K-chunk into LDS buffer `buf`
    //      32 x global_load_async_to_lds_b128 per thread per chunk
    auto stage_async = [&](int k0, int buf) {
        for (int i = tid; i < TOK_PER_WG * (KC / 4); i += 128) {
            const int r  = i >> 5;              // token row
            const int cq = (i & 31) << 2;       // float column
            const float*   src = &x[(size_t)(tok0 + r) * DIM + k0 + cq];
            const unsigned dst = (unsigned)(uintptr_t)&xs[buf][r * KSTRIDE + cq];
            asm volatile("global_load_async_to_lds_b128 %0, %1, off"
                         :: "v"(dst), "v"(src) : "memory");
        }
        for (int i = tid; i < NEXP * (KC / 4); i += 128) {
            const int r  = i >> 5;              // expert row
            const int cq = (i & 31) << 2;
            const float*   src = &w[(size_t)r * DIM + k0 + cq];
            const unsigned dst = (unsigned)(uintptr_t)&ws2[buf][r * KSTRIDE + cq];
            asm volatile("global_load_async_to_lds_b128 %0, %1, off"
                         :: "v"(dst), "v"(src) : "memory");
        }
    };

    // 16 tokens x 64 experts per wave: 4 accumulator tiles
    v8f c0 = {}, c1 = {}, c2 = {}, c3 = {};

    stage_async(0, 0);
    int buf = 0;

    for (int k0 = 0; k0 < DIM; k0 += KC) {
        if (k0 + KC < DIM) {
            stage_async(k0 + KC, buf ^ 1);                  // prefetch next chunk
            asm volatile("s_wait_asynccnt 0x20" ::: "memory");  // chunk k0 done
        } else {
            asm volatile("s_wait_asynccnt 0x0" ::: "memory");
        }
        __syncthreads();   // all waves' async writes visible

        // ---- WMMA over the chunk: A = 16x4 fp32, B = 4x16 fp32 per tile
        const float* xrow = &xs[buf][(wave * 16 + row16) * KSTRIDE];
        const float* w0   = &ws2[buf][( 0 + row16) * KSTRIDE];
        const float* w1   = &ws2[buf][(16 + row16) * KSTRIDE];
        const float* w2   = &ws2[buf][(32 + row16) * KSTRIDE];
        const float* w3   = &ws2[buf][(48 + row16) * KSTRIDE];

#pragma unroll 4
        for (int kk = 0; kk < KC; kk += 4) {
            const int kc = kk + half * 2;  // lanes 0-15: K0,K1; lanes 16-31: K2,K3
            const v2f a  = *reinterpret_cast<const v2f*>(xrow + kc);
            const v2f b0 = *reinterpret_cast<const v2f*>(w0 + kc);
            const v2f b1 = *reinterpret_cast<const v2f*>(w1 + kc);
            const v2f b2 = *reinterpret_cast<const v2f*>(w2 + kc);
            const v2f b3 = *reinterpret_cast<const v2f*>(w3 + kc);
            c0 = __builtin_amdgcn_wmma_f32_16x16x4_f32(false, a, false, b0, (short)0, c0, false, false);
            c1 = __builtin_amdgcn_wmma_f32_16x16x4_f32(false, a, false, b1, (short)0, c1, false, false);
            c2 = __builtin_amdgcn_wmma_f32_16x16x4_f32(false, a, false, b2, (short)0, c2, false, false);
            c3 = __builtin_amdgcn_wmma_f32_16x16x4_f32(false, a, false, b3, (short)0, c3, false, false);
        }

        __syncthreads();   // done reading buf before it is overwritten
        buf ^= 1;
    }

    // ---- spill logits to LDS: lane L, VGPR r holds (M = r + 8*half, N = row16)
#pragma unroll
    for (int r = 0; r < 8; ++r) {
        const int trow = wave * 16 + half * 8 + r;
        lprob[trow * LSTRIDE +  0 + row16] = c0[r];
        lprob[trow * LSTRIDE + 16 + row16] = c1[r];
        lprob[trow * LSTRIDE + 32 + row16] = c2[r];
        lprob[trow * LSTRIDE + 48 + row16] = c3[r];
    }
    __syncthreads();

    // ---- phase 2: one thread per token -> softmax, top-8, normalize, emit
    if (tid < TOK_PER_WG) {
        float* row = &lprob[tid * LSTRIDE];
        float mx = row[0];
        for (int e = 1; e < NEXP; ++e) mx = fmaxf(mx, row[e]);
        float s = 0.0f;
        for (int e = 0; e < NEXP; ++e) {
            const float p = expf(row[e] - mx);
            row[e] = p;
            s += p;
        }
        const float inv = 1.0f / s;

        unsigned long long used = 0ull;
        float topv[TOPK];
        int   topi[TOPK];
        float wsum = 0.0f;
        for (int j = 0; j < TOPK; ++j) {
            float best = -1.0f;
            int   bi   = 0;
            for (int e = 0; e < NEXP; ++e) {
                if (!((used >> e) & 1ull)) {
                    const float pv = row[e];
                    if (pv > best) { best = pv; bi = e; }  // ties -> lowest index
                }
            }
            used |= (1ull << bi);
            const float prob = best * inv;
            topv[j] = prob;
            topi[j] = bi;
            wsum += prob;
            atomicAdd(&lcount[bi], 1u);
        }
        const float invw = 1.0f / (wsum + 1e-20f);
        const size_t o = (size_t)(tok0 + tid) * TOPK;
        for (int j = 0; j < TOPK; ++j) {
            out_idx[o + j] = (float)topi[j];
            out_w[o + j]   = topv[j] * invw;
        }
        // leave normalized probs in LDS for the column reduction
        for (int e = 0; e < NEXP; ++e) row[e] *= inv;
    }
    __syncthreads();

    // ---- phase 3: per-expert column sums for aux loss, one global atomic each
    if (tid < NEXP) {
        float cs = 0.0f;
        for (int t = 0; t < TOK_PER_WG; ++t) cs += lprob[t * LSTRIDE + tid];
        const int b = tok0 / SEQ;  // WG never crosses a batch boundary
        atomicAdd(&g_score_sum[b * NEXP + tid], cs);
        atomicAdd(&g_count[b * NEXP + tid], (float)lcount[tid]);
    }
}

// ---------------------------------------------------------------------------
__global__ void moe_ws_init_kernel(float* ws) {
    ws[threadIdx.x] = 0.0f;   // 512 floats: score_sum + count
}

__global__ void moe_aux_kernel(const float* __restrict__ g_score_sum,
                               const float* __restrict__ g_count,
                               float* __restrict__ out_aux)
{
    __shared__ float red[BATCH * NEXP];
    const int tid = threadIdx.x;  // 0..255 -> (b, e) cell
    const float ce = g_count[tid] * ((float)NEXP / (float)(SEQ * TOPK));
    const float sm = g_score_sum[tid] / (float)SEQ;
    red[tid] = ce * sm;
    __syncthreads();
    for (int s = (BATCH * NEXP) / 2; s > 0; s >>= 1) {
        if (tid < s) red[tid] += red[tid + s];
        __syncthreads();
    }
    if (tid == 0) *out_aux = red[0] * (ALPHA_F / (float)BATCH);
}

// ---------------------------------------------------------------------------
extern "C" void kernel_launch(void* const* d_in, const int* in_sizes, int n_in,
                              void* d_out, int out_size, void* d_ws, size_t ws_size,
                              hipStream_t stream)
{
    const float* x = (const float*)d_in[0];   // [4, 4096, 2048] fp32
    const float* w = (const float*)d_in[1];   // [64, 2048] fp32

    float* out      = (float*)d_out;
    float* out_idx  = out;                                    // T*K
    float* out_w    = out + (size_t)T_TOKENS * TOPK;          // T*K
    float* out_aux  = out + 2 * (size_t)T_TOKENS * TOPK;      // 1

    float* g_score  = (float*)d_ws;                           // [4,64]
    float* g_count  = g_score + BATCH * NEXP;                 // [4,64]

    moe_ws_init_kernel<<<1, 2 * BATCH * NEXP, 0, stream>>>(g_score);
    moe_gate_kernel<<<T_TOKENS / TOK_PER_WG, 128, 0, stream>>>(
        x, w, out_idx, out_w, g_score, g_count);
    moe_aux_kernel<<<1, BATCH * NEXP, 0, stream>>>(g_score, g_count, out_aux);
}